// modeler_warm_18571438588577
// MI455X (gfx1250) — compile-verified
//
#include <hip/hip_runtime.h>
#include <hip/hip_bf16.h>

#define NN 50000
#define EE 300000
#define DD 256
#define FF 1024
#define CC 40

typedef float v2f __attribute__((ext_vector_type(2)));
typedef float v8f __attribute__((ext_vector_type(8)));

// ---------------------------------------------------------------------------
// Degree counting (integer atomics, exact)
// ---------------------------------------------------------------------------
__global__ void deg_kernel(const int* __restrict__ src, const int* __restrict__ dst,
                           int* __restrict__ degO, int* __restrict__ degI, int E) {
    int e = blockIdx.x * blockDim.x + threadIdx.x;
    if (e >= E) return;
    atomicAdd(&degO[src[e]], 1);
    atomicAdd(&degI[dst[e]], 1);
}

__global__ void norm_kernel(const int* __restrict__ degO, const int* __restrict__ degI,
                            float* __restrict__ ns, float* __restrict__ nd, int N) {
    int i = blockIdx.x * blockDim.x + threadIdx.x;
    if (i >= N) return;
    int dO = degO[i]; if (dO < 1) dO = 1;
    int dI = degI[i]; if (dI < 1) dI = 1;
    ns[i] = rsqrtf((float)dO);
    nd[i] = rsqrtf((float)dI);
}

// ---------------------------------------------------------------------------
// WMMA fp32 GEMM:  C[M,Nc] = (A[M,K] * rowscale?) @ B[K,Nc] + bias[Nc]
// One wave computes a 16x64 tile of C (4 accumulators) via
// v_wmma_f32_16x16x4_f32. K / Nc / nTilesN are compile-time so all B loads
// use immediate IOFFSETs off a single pointer bumped once per k-step.
// A frag (16x4 f32): lanes 0-15 -> M=lane, K={k,k+1}; lanes 16-31 -> K={k+2,k+3}
// B frag (4x16 f32): lanes 0-15 -> N=lane, rows {k,k+1}; lanes 16-31 -> {k+2,k+3}
// C frag: VGPR r -> row r + 8*(lane>>4), col = lane&15
// ---------------------------------------------------------------------------
template<bool SCALE_A, bool COL_GUARD, int K, int Nc, int NTILES>
__global__ void gemm_wmma_kernel(const float* __restrict__ A,
                                 const float* __restrict__ rowscale,
                                 const float* __restrict__ B,
                                 const float* __restrict__ bias,
                                 float* __restrict__ C,
                                 int M) {
    int wid  = (blockIdx.x * blockDim.x + threadIdx.x) >> 5;
    int lane = threadIdx.x & 31;
    int tileM = wid / NTILES;               // compile-time NTILES -> shift
    int tileN = wid - tileM * NTILES;
    int mBase = tileM << 4;
    if (mBase >= M) return;                 // wave-uniform exit (EXEC all-1 for WMMA)
    int nBase = tileN << 6;                 // 64-wide wave tile

    int lo = lane & 15;
    int hi = lane >> 4;                     // 0 or 1 -> K sub-pair
    int aRowIdx = mBase + lo;               // A row for this lane
    int col0 = nBase + lo;                  // subtile columns: col0 + 16*j

    float scale = SCALE_A ? rowscale[aRowIdx] : 1.0f;

    bool ok0 = (!COL_GUARD) || (col0      < Nc);
    bool ok1 = (!COL_GUARD) || (col0 + 16 < Nc);
    bool ok2 = (!COL_GUARD) || (col0 + 32 < Nc);
    bool ok3 = (!COL_GUARD) || (col0 + 48 < Nc);

    const float* aP = A + (size_t)aRowIdx * K + (hi << 1);
    const float* bP = B + (size_t)(hi << 1) * Nc + col0;

    v8f acc0 = {}, acc1 = {}, acc2 = {}, acc3 = {};
#pragma unroll 2
    for (int k = 0; k < K; k += 4) {
        float2 av = *(const float2*)aP;
        aP += 4;
        v2f a;
        a.x = av.x; a.y = av.y;
        if (SCALE_A) { a.x *= scale; a.y *= scale; }

        v2f b0, b1, b2, b3;
        if (ok0) { b0.x = bP[0];  b0.y = bP[Nc];      } else { b0.x = 0.f; b0.y = 0.f; }
        if (ok1) { b1.x = bP[16]; b1.y = bP[Nc + 16]; } else { b1.x = 0.f; b1.y = 0.f; }
        if (ok2) { b2.x = bP[32]; b2.y = bP[Nc + 32]; } else { b2.x = 0.f; b2.y = 0.f; }
        if (ok3) { b3.x = bP[48]; b3.y = bP[Nc + 48]; } else { b3.x = 0.f; b3.y = 0.f; }
        bP += 4 * Nc;

        acc0 = __builtin_amdgcn_wmma_f32_16x16x4_f32(false, a, false, b0, (short)0, acc0, false, false);
        acc1 = __builtin_amdgcn_wmma_f32_16x16x4_f32(false, a, false, b1, (short)0, acc1, false, false);
        acc2 = __builtin_amdgcn_wmma_f32_16x16x4_f32(false, a, false, b2, (short)0, acc2, false, false);
        acc3 = __builtin_amdgcn_wmma_f32_16x16x4_f32(false, a, false, b3, (short)0, acc3, false, false);
    }

    // Store 4 subtiles (guarded per subtile/lane for the Nc=40 classifier)
    int storeRow = mBase + (hi << 3);
#pragma unroll
    for (int j = 0; j < 4; ++j) {
        bool ok = (j == 0) ? ok0 : (j == 1) ? ok1 : (j == 2) ? ok2 : ok3;
        if (!ok) continue;
        int c = col0 + (j << 4);
        float bv = bias[c];
        const v8f& acc = (j == 0) ? acc0 : (j == 1) ? acc1 : (j == 2) ? acc2 : acc3;
#pragma unroll
        for (int r = 0; r < 8; ++r) {
            C[(size_t)(storeRow + r) * Nc + c] = acc[r] + bv;
        }
    }
}

// ---------------------------------------------------------------------------
// SpMM scatter:  m[dst[e], :] += x[src[e], :] * norm_s[src[e]]
// 64 threads per edge, float4 per thread, hardware float atomics.
// ---------------------------------------------------------------------------
__global__ void spmm_kernel(const float* __restrict__ x, const float* __restrict__ norm_s,
                            const int* __restrict__ src, const int* __restrict__ dst,
                            float* __restrict__ m, int E) {
    long long t = (long long)blockIdx.x * blockDim.x + threadIdx.x;
    int e = (int)(t >> 6);
    if (e >= E) return;
    int d = ((int)t & 63) << 2;
    int s  = src[e];
    int dn = dst[e];
    float ns = norm_s[s];
    float4 v = *(const float4*)(x + (size_t)s * DD + d);
    float* out = m + (size_t)dn * DD + d;
    unsafeAtomicAdd(out + 0, v.x * ns);
    unsafeAtomicAdd(out + 1, v.y * ns);
    unsafeAtomicAdd(out + 2, v.z * ns);
    unsafeAtomicAdd(out + 3, v.w * ns);
}

// ---------------------------------------------------------------------------
// BatchNorm column stats: thread t = column t (coalesced row reads),
// grid-stride over rows, float-atomic partial reduction.
// ---------------------------------------------------------------------------
__global__ void bn_stat_kernel(const float* __restrict__ x,
                               float* __restrict__ sum, float* __restrict__ sumsq,
                               int Nrows) {
    int col = threadIdx.x;            // blockDim.x == DD
    float s = 0.0f, q = 0.0f;
    for (int r = blockIdx.x; r < Nrows; r += gridDim.x) {
        float v = x[(size_t)r * DD + col];
        s += v;
        q += v * v;
    }
    unsafeAtomicAdd(&sum[col], s);
    unsafeAtomicAdd(&sumsq[col], q);
}

__global__ void bn_fin_kernel(const float* __restrict__ sum, const float* __restrict__ sumsq,
                              float* __restrict__ mu, float* __restrict__ rstd, int Nrows) {
    int c = threadIdx.x;              // blockDim.x == DD
    float invN = 1.0f / (float)Nrows;
    float m = sum[c] * invN;
    float var = sumsq[c] * invN - m * m;
    mu[c] = m;
    rstd[c] = rsqrtf(var + 1e-5f);
}

__global__ void bn_elu_kernel(float* __restrict__ x,
                              const float* __restrict__ mu, const float* __restrict__ rstd,
                              const float* __restrict__ gamma, const float* __restrict__ beta,
                              long long total) {
    long long i = (long long)blockIdx.x * blockDim.x + threadIdx.x;
    if (i >= total) return;
    int col = (int)(i & (DD - 1));
    float y = gamma[col] * (x[i] - mu[col]) * rstd[col] + beta[col];
    x[i] = (y > 0.0f) ? y : expm1f(y);
}

// ---------------------------------------------------------------------------
// Host launcher
// ---------------------------------------------------------------------------
extern "C" void kernel_launch(void* const* d_in, const int* in_sizes, int n_in,
                              void* d_out, int out_size, void* d_ws, size_t ws_size,
                              hipStream_t stream) {
    // Input order: feat, src, dst, W_fc, b_fc, W1, b1, W2, b2, W3, b3, gamma, beta, W_lin, b_lin
    const float* feat  = (const float*)d_in[0];
    const int*   src   = (const int*)  d_in[1];
    const int*   dst   = (const int*)  d_in[2];
    const float* W_fc  = (const float*)d_in[3];
    const float* b_fc  = (const float*)d_in[4];
    const float* W1    = (const float*)d_in[5];
    const float* b1    = (const float*)d_in[6];
    const float* W2    = (const float*)d_in[7];
    const float* b2    = (const float*)d_in[8];
    const float* W3    = (const float*)d_in[9];
    const float* b3    = (const float*)d_in[10];
    const float* gamma = (const float*)d_in[11];
    const float* beta  = (const float*)d_in[12];
    const float* W_lin = (const float*)d_in[13];
    const float* b_lin = (const float*)d_in[14];

    float* out_x     = (float*)d_out;                       // [N, D]
    float* out_logit = (float*)d_out + (size_t)NN * DD;     // [N, C]

    // Workspace layout
    float* xA     = (float*)d_ws;                 // N*D
    float* xB     = xA + (size_t)NN * DD;         // N*D (SpMM accumulator)
    float* norm_s = xB + (size_t)NN * DD;         // N
    float* norm_d = norm_s + NN;                  // N
    int*   degO   = (int*)(norm_d + NN);          // N
    int*   degI   = degO + NN;                    // N
    float* bnsum  = (float*)(degI + NN);          // D
    float* bnsq   = bnsum + DD;                   // D
    float* mu     = bnsq + DD;                    // D
    float* rstd   = mu + DD;                      // D

    const int TB = 256;

    // --- degrees & norms ---
    hipMemsetAsync(degO, 0, 2 * (size_t)NN * sizeof(int), stream);
    deg_kernel<<<(EE + TB - 1) / TB, TB, 0, stream>>>(src, dst, degO, degI, EE);
    norm_kernel<<<(NN + TB - 1) / TB, TB, 0, stream>>>(degO, degI, norm_s, norm_d, NN);

    // GEMM launch geometry: 8 waves per 256-thread block, one 16x64 tile per wave
    auto gemmBlocks = [](int M, int nTilesN) {
        long long waves = (long long)((M + 15) / 16) * nTilesN;
        return (int)((waves + 7) / 8);
    };
    const int tilesD = DD / 64;             // 4 wave-tiles across 256 cols
    const int tilesC = (CC + 63) / 64;      // 1 (guarded) wave-tile across 40 cols

    // --- x = feat @ W_fc + b_fc ---
    gemm_wmma_kernel<false, false, FF, DD, 4><<<gemmBlocks(NN, tilesD), TB, 0, stream>>>(
        feat, nullptr, W_fc, b_fc, xA, NN);

    long long spmmThreads = (long long)EE * 64;
    int spmmBlocks = (int)((spmmThreads + TB - 1) / TB);
    long long total = (long long)NN * DD;
    int eluBlocks = (int)((total + TB - 1) / TB);

    // ===== Layer 1 =====
    hipMemsetAsync(xB, 0, (size_t)NN * DD * sizeof(float), stream);
    spmm_kernel<<<spmmBlocks, TB, 0, stream>>>(xA, norm_s, src, dst, xB, EE);
    gemm_wmma_kernel<true, false, DD, DD, 4><<<gemmBlocks(NN, tilesD), TB, 0, stream>>>(
        xB, norm_d, W1, b1, xA, NN);
    hipMemsetAsync(bnsum, 0, 2 * DD * sizeof(float), stream);
    bn_stat_kernel<<<512, DD, 0, stream>>>(xA, bnsum, bnsq, NN);
    bn_fin_kernel<<<1, DD, 0, stream>>>(bnsum, bnsq, mu, rstd, NN);
    bn_elu_kernel<<<eluBlocks, TB, 0, stream>>>(xA, mu, rstd, gamma, beta, total);

    // ===== Layer 2 =====
    hipMemsetAsync(xB, 0, (size_t)NN * DD * sizeof(float), stream);
    spmm_kernel<<<spmmBlocks, TB, 0, stream>>>(xA, norm_s, src, dst, xB, EE);
    gemm_wmma_kernel<true, false, DD, DD, 4><<<gemmBlocks(NN, tilesD), TB, 0, stream>>>(
        xB, norm_d, W2, b2, xA, NN);
    hipMemsetAsync(bnsum, 0, 2 * DD * sizeof(float), stream);
    bn_stat_kernel<<<512, DD, 0, stream>>>(xA, bnsum, bnsq, NN);
    bn_fin_kernel<<<1, DD, 0, stream>>>(bnsum, bnsq, mu, rstd, NN);
    bn_elu_kernel<<<eluBlocks, TB, 0, stream>>>(xA, mu, rstd, gamma, beta, total);

    // ===== Layer 3 (writes straight into d_out x) =====
    hipMemsetAsync(xB, 0, (size_t)NN * DD * sizeof(float), stream);
    spmm_kernel<<<spmmBlocks, TB, 0, stream>>>(xA, norm_s, src, dst, xB, EE);
    gemm_wmma_kernel<true, false, DD, DD, 4><<<gemmBlocks(NN, tilesD), TB, 0, stream>>>(
        xB, norm_d, W3, b3, out_x, NN);

    // ===== Classifier: logits = x @ W_lin + b_lin (Nc = 40, guarded) =====
    gemm_wmma_kernel<false, true, DD, CC, 1><<<gemmBlocks(NN, tilesC), TB, 0, stream>>>(
        out_x, nullptr, W_lin, b_lin, out_logit, NN);
}